// Model_32779190403171
// MI455X (gfx1250) — compile-verified
//
#include <hip/hip_runtime.h>
#include <hip/hip_bf16.h>
#include <math.h>

// ---------------------------------------------------------------------------
// Sobel magnitude+angle, 8192x8192 f32, for MI455X (gfx1250, wave32).
//
// Memory-bound: 768MB traffic / 23.3 TB/s ~ 33us floor. Strategy:
//  * Stage each block's 34x68 halo tile into LDS with CDNA5 async
//    global->LDS copies (ASYNCcnt path), read each input byte ~1.13x.
//  * Horizontal 3-tap convolution done as a banded matmul on the matrix
//    pipe: V_WMMA_F32_16X16X4_F32, D(16x16) = A(16x20) x B(20x16),
//    5 K-chunks per gradient. Vertical 3-tap reduction is fused into the
//    A-operand build (separable Sobel). WMMA co-executes with the VALU,
//    which is busy with atan2f/sqrtf (the real VALU cost here).
// ---------------------------------------------------------------------------

typedef float v2f __attribute__((ext_vector_type(2)));
typedef float v8f __attribute__((ext_vector_type(8)));

#define IMG_H 8192
#define IMG_W 8192

// Workgroup output tile: 64 (cols) x 32 (rows) = 8 waves x (16x16 WMMA tiles)
#define TILE_W 64
#define TILE_H 32
// LDS staging tile with halo + K-padding: rows 34, cols 68 (ox up to 48 + K=20)
#define LDS_H 34
#define LDS_W 68
#define LDS_N (LDS_H * LDS_W)

__global__ __launch_bounds__(256)
void sobel_wmma_kernel(const float* __restrict__ image, float* __restrict__ out)
{
    __shared__ float tile[LDS_N];

    const int tid    = threadIdx.x;
    const int lane   = tid & 31;          // wave32
    const int wid    = tid >> 5;          // 8 waves per block
    const int wgRow0 = blockIdx.y * TILE_H;
    const int wgCol0 = blockIdx.x * TILE_W;

    // ---------------- Stage input tile (global -> LDS, async) ----------------
    // Halo origin (wgRow0-1, wgCol0-1); interior blocks need no zero-fill.
    const bool interior = (wgRow0 >= 1) && (wgRow0 + (LDS_H - 1) <= IMG_H) &&
                          (wgCol0 >= 1) && (wgCol0 + (LDS_W - 1) <= IMG_W);

    if (!interior) {
        // Zero padding (reference uses zero-padded conv). Edge blocks only.
        for (int idx = tid; idx < LDS_N; idx += 256) tile[idx] = 0.0f;
        __syncthreads();
    }

    for (int idx = tid; idx < LDS_N; idx += 256) {
        const int r  = idx / LDS_W;
        const int c  = idx - r * LDS_W;
        const int gr = wgRow0 - 1 + r;
        const int gc = wgCol0 - 1 + c;
        if (interior || ((unsigned)gr < (unsigned)IMG_H &&
                         (unsigned)gc < (unsigned)IMG_W)) {
            // LDS flat-address low 32 bits == LDS byte offset (ISA aperture rule).
            unsigned int       lb = (unsigned int)(uintptr_t)(&tile[idx]);
            unsigned long long ga = (unsigned long long)(uintptr_t)
                                    (image + (size_t)gr * IMG_W + gc);
            asm volatile("global_load_async_to_lds_b32 %0, %1, off"
                         :: "v"(lb), "v"(ga) : "memory");
        }
    }
    asm volatile("s_wait_asynccnt 0" ::: "memory");
    __syncthreads();

    // ---------------- Per-wave 16x16 WMMA tile ----------------
    const int ox = (wid & 3) * 16;   // tile col origin inside block tile
    const int oy = (wid >> 2) * 16;  // tile row origin inside block tile

    // f32 16x16x4 A layout (ISA 7.12.2): lanes 0-15 hold K={0,1}, lanes 16-31
    // hold K={2,3}, M = lane&15 striped down rows. B assumed mirrored
    // (VGPR0: K=0 / K=2 across half-waves, VGPR1: K=1 / K=3), N = lane&15.
    const int m     = lane & 15;
    const int ksel2 = (lane >> 4) << 1;   // 0 or 2
    const int n     = lane & 15;          // B/C/D column

    v8f cgx = {};  // gx accumulator (16x16 f32 C/D)
    v8f cgy = {};  // gy accumulator

    #pragma unroll
    for (int kc = 0; kc < 20; kc += 4) {
        // ---- A operands: vertical Sobel reduction fused in (separable) ----
        // padded col k maps to global col wgCol0 + k - 1; rows m..m+2 give the
        // 3-row vertical window centered on output row wgRow0 + oy + m.
        const float* rb = &tile[(oy + m) * LDS_W + (ox + kc + ksel2)];
        const float x00 = rb[0],           x01 = rb[1];
        const float x10 = rb[LDS_W],       x11 = rb[LDS_W + 1];
        const float x20 = rb[2 * LDS_W],   x21 = rb[2 * LDS_W + 1];
        v2f As, Ad;
        As.x = x00 + 2.0f * x10 + x20;  As.y = x01 + 2.0f * x11 + x21;  // [1,2,1]^T
        Ad.x = x20 - x00;               Ad.y = x21 - x01;               // [-1,0,1]^T

        // ---- B operands: banded horizontal taps B[k][n] = h[k-n] ----
        // hx = [-1,0,1] (gx), hy = [1,2,1] (gy); k=18,19 rows are zero pad.
        const int e0 = kc + ksel2 - n;
        const int e1 = e0 + 1;
        v2f Bx, By;
        Bx.x = (e0 == 0) ? -1.0f : (e0 == 2) ? 1.0f : 0.0f;
        Bx.y = (e1 == 0) ? -1.0f : (e1 == 2) ? 1.0f : 0.0f;
        By.x = (e0 == 1) ? 2.0f : (e0 == 0 || e0 == 2) ? 1.0f : 0.0f;
        By.y = (e1 == 1) ? 2.0f : (e1 == 0 || e1 == 2) ? 1.0f : 0.0f;

        // 8-arg form: (neg_a, A, neg_b, B, c_mod, C, reuse_a, reuse_b)
        cgx = __builtin_amdgcn_wmma_f32_16x16x4_f32(
                  false, As, false, Bx, (short)0, cgx, false, false);
        cgy = __builtin_amdgcn_wmma_f32_16x16x4_f32(
                  false, Ad, false, By, (short)0, cgy, false, false);
    }

    // ---------------- Epilogue: magnitude + angle, coalesced stores ----------
    // C/D layout: VGPR i -> M = i (lanes 0-15) or M = 8+i (lanes 16-31), N = lane&15.
    const int    mbase = (lane >> 4) * 8;
    const size_t plane = (size_t)IMG_H * IMG_W;
    #pragma unroll
    for (int i = 0; i < 8; ++i) {
        const float gxv = cgx[i];
        const float gyv = cgy[i];
        const float mag = __builtin_sqrtf(gxv * gxv + gyv * gyv);
        const float ang = atan2f(gyv, gxv);
        const size_t o  = (size_t)(wgRow0 + oy + mbase + i) * IMG_W
                        + (size_t)(wgCol0 + ox + n);
        out[o]         = mag;
        out[plane + o] = ang;
    }
}

extern "C" void kernel_launch(void* const* d_in, const int* in_sizes, int n_in,
                              void* d_out, int out_size, void* d_ws, size_t ws_size,
                              hipStream_t stream) {
    (void)in_sizes; (void)n_in; (void)d_ws; (void)ws_size; (void)out_size;
    // d_in[0] = image (f32 8192x8192); d_in[1]/d_in[2] = sobel_x/sobel_y
    // (compile-time constants in the reference; taps are folded into the
    // banded B matrices above).
    const float* image = (const float*)d_in[0];
    float*       out   = (float*)d_out;  // [magnitude plane | angle plane]

    dim3 grid(IMG_W / TILE_W, IMG_H / TILE_H);  // 128 x 256
    dim3 block(256);                            // 8 wave32 waves
    sobel_wmma_kernel<<<grid, block, 0, stream>>>(image, out);
}